// MVDR_11587821765059
// MI455X (gfx1250) — compile-verified
//
#include <hip/hip_runtime.h>

// MVDR beamformer, fully fused: one workgroup per (b,f).
//   X tile (6x1000 complex fp32) staged once in LDS (async global->LDS DMA,
//   ASYNCcnt path) -> HBM traffic ~263MB total => memory-bound, ~11us at 23.3TB/s.
//   PSDs computed as 16x16 (12x12 useful) real WMMA rank-K updates
//   (V_WMMA_F32_16X16X4_F32, full fp32), 6x6 complex Gauss-Jordan on one lane,
//   beamform from the LDS copy.

typedef float v2f __attribute__((ext_vector_type(2)));
typedef float v8f __attribute__((ext_vector_type(8)));
typedef int   v4i __attribute__((ext_vector_type(4)));

#define Bn 4
#define Fn 1025
#define Cn 6
#define Tn 1000
// Row stride in floats. 1028*4 = 4112 bytes (16B aligned for b128 async DMA).
// Bank for (row,t) = (4*row + t) % 64: rows 0..15 hit 16 distinct banks, and
// the upper half-wave (t+2) cannot alias a multiple-of-4 offset -> conflict-free.
#define XSTRIDE 1028

// LDS layout (float offsets)
#define OFF_X     0                       // 16 rows x XSTRIDE (0-5 re, 6-11 im, 12-15 zero)
#define OFF_MS    (16 * XSTRIDE)          // normalized mask_s, Tn floats (16B aligned)
#define OFF_MN    (OFF_MS + Tn)           // normalized mask_n, Tn floats (16B aligned)
#define OFF_PART  (OFF_MN + Tn)           // 8 waves * 512 partial PSD floats
#define OFF_PRED  (OFF_PART + 8 * 512)    // reduced P_s | P_n (2*256)
#define OFF_W     (OFF_PRED + 512)        // w vector, 6 complex = 12 floats
#define OFF_SUM   (OFF_W + 12)            // mask sums (2)
#define SMEM_FLOATS (OFF_SUM + 2)

#if __has_builtin(__builtin_amdgcn_global_load_async_to_lds_b128)
#define HAVE_ASYNC_LDS 1
#else
#define HAVE_ASYNC_LDS 0
#endif

#if HAVE_ASYNC_LDS
__device__ __forceinline__ void async_cp16(void* lds, const void* g) {
  __builtin_amdgcn_global_load_async_to_lds_b128(
      (__attribute__((address_space(1))) v4i*)g,
      (__attribute__((address_space(3))) v4i*)lds,
      0 /*offset*/, 0 /*cpol*/);
}
#endif

// D-matrix (16x16 f32, wave32) flat index for element (M,N):
// VGPR j = M&7, lane = (M>>3)*16 + N  ->  flat = j*32 + lane
__device__ __forceinline__ int pidx(int M, int N) {
  return ((M & 7) << 5) + ((M >> 3) << 4) + N;
}

__global__ __launch_bounds__(256) void mvdr_kernel(
    const float* __restrict__ sr, const float* __restrict__ si,
    const float* __restrict__ ms, const float* __restrict__ mn,
    float* __restrict__ out)
{
  __shared__ float sm[SMEM_FLOATS];
  const int bf   = blockIdx.x;
  const int tid  = threadIdx.x;
  const int lane = tid & 31;                                     // wave32
  const int wave = __builtin_amdgcn_readfirstlane(tid >> 5);     // scalar wave id

  const float* srb = sr + (size_t)bf * Cn * Tn;
  const float* sib = si + (size_t)bf * Cn * Tn;
  const float* msb = ms + (size_t)bf * Tn;
  const float* mnb = mn + (size_t)bf * Tn;

  // ---------------- Phase A: stage X + masks into LDS (single HBM read of X)
#if HAVE_ASYNC_LDS
  // 16-byte async DMA chunks, no VGPR round-trip (ASYNCcnt path).
  for (int idx = tid; idx < 12 * (Tn / 4); idx += 256) {         // X: 12 rows * 250 chunks
    int r = idx / (Tn / 4), ch = idx - r * (Tn / 4);
    const float* g = (r < 6) ? (srb + r * Tn + ch * 4) : (sib + (r - 6) * Tn + ch * 4);
    async_cp16(&sm[OFF_X + r * XSTRIDE + ch * 4], g);
  }
  for (int idx = tid; idx < 2 * (Tn / 4); idx += 256) {          // masks: 2 * 250 chunks
    int which = idx / (Tn / 4), ch = idx - which * (Tn / 4);
    const float* g = which ? (mnb + ch * 4) : (msb + ch * 4);
    async_cp16(&sm[(which ? OFF_MN : OFF_MS) + ch * 4], g);
  }
#else
  for (int idx = tid; idx < Cn * Tn; idx += 256) {
    int c = idx / Tn, t = idx - c * Tn;
    sm[OFF_X + c * XSTRIDE + t]       = srb[idx];
    sm[OFF_X + (6 + c) * XSTRIDE + t] = sib[idx];
  }
  for (int t = tid; t < Tn; t += 256) {
    sm[OFF_MS + t] = msb[t];
    sm[OFF_MN + t] = mnb[t];
  }
#endif
  for (int idx = tid; idx < 4 * Tn; idx += 256) {                // zero pad rows 12..15
    int r = 12 + idx / Tn, t = idx % Tn;
    sm[OFF_X + r * XSTRIDE + t] = 0.f;
  }
  if (tid < 2) sm[OFF_SUM + tid] = 0.f;
#if HAVE_ASYNC_LDS
  asm volatile("s_wait_asynccnt 0" ::: "memory");
#endif
  __syncthreads();

  // ---------------- mask normalization: m / (sum_t m + 1e-15)
  {
    float ls = 0.f, ln = 0.f;
    for (int t = tid; t < Tn; t += 256) { ls += sm[OFF_MS + t]; ln += sm[OFF_MN + t]; }
    atomicAdd(&sm[OFF_SUM + 0], ls);
    atomicAdd(&sm[OFF_SUM + 1], ln);
  }
  __syncthreads();
  {
    float invs = 1.f / (sm[OFF_SUM + 0] + 1e-15f);
    float invn = 1.f / (sm[OFF_SUM + 1] + 1e-15f);
    for (int t = tid; t < Tn; t += 256) {
      sm[OFF_MS + t] *= invs;
      sm[OFF_MN + t] *= invn;
    }
  }
  __syncthreads();

  // ---------------- Phase B: PSDs via V_WMMA_F32_16X16X4_F32 rank-K updates
  // A (16x4): lanes 0-15 row M=lane hold K=0,1; lanes 16-31 row M=lane-16 hold K=2,3.
  // B (4x16): lanes index N=lane%16; v0/v1 hold K = {0,1} (lanes<16) / {2,3} (lanes>=16).
  // B is A's data scaled by the normalized mask -> identical LDS addresses.
  v8f accS = {0.f, 0.f, 0.f, 0.f, 0.f, 0.f, 0.f, 0.f};
  v8f accN = {0.f, 0.f, 0.f, 0.f, 0.f, 0.f, 0.f, 0.f};
  {
    const int row    = lane & 15;
    const int koff   = (lane >> 4) << 1;                // 0 or 2
    const int nsteps = ((Tn / 4) - wave + 7) >> 3;      // scalar trip count (31 or 32)
    int t0 = wave * 4 + koff;
    for (int i = 0; i < nsteps; ++i, t0 += 32) {
      float x0 = sm[OFF_X + row * XSTRIDE + t0];
      float x1 = sm[OFF_X + row * XSTRIDE + t0 + 1];
      float s0 = sm[OFF_MS + t0], s1 = sm[OFF_MS + t0 + 1];
      float n0 = sm[OFF_MN + t0], n1 = sm[OFF_MN + t0 + 1];
      v2f a;   a.x   = x0;      a.y   = x1;
      v2f bs;  bs.x  = s0 * x0; bs.y  = s1 * x1;
      v2f bn_; bn_.x = n0 * x0; bn_.y = n1 * x1;
      accS = __builtin_amdgcn_wmma_f32_16x16x4_f32(false, a, false, bs,  (short)0, accS, false, false);
      accN = __builtin_amdgcn_wmma_f32_16x16x4_f32(false, a, false, bn_, (short)0, accN, false, false);
    }
  }
  // per-wave partials -> LDS
#pragma unroll
  for (int j = 0; j < 8; j++) {
    sm[OFF_PART + wave * 512 +       j * 32 + lane] = accS[j];
    sm[OFF_PART + wave * 512 + 256 + j * 32 + lane] = accN[j];
  }
  __syncthreads();
  // reduce 8 waves -> Pred[0..255]=P_s, Pred[256..511]=P_n
  for (int p = tid; p < 512; p += 256) {
    float acc = 0.f;
#pragma unroll
    for (int w = 0; w < 8; w++) acc += sm[OFF_PART + w * 512 + p];
    sm[OFF_PRED + p] = acc;
  }
  __syncthreads();

  // ---------------- Phase C: 6x6 complex solve (serial, lane 0)
  if (tid == 0) {
    float Ar[6][12], Ai[6][12];   // augmented [psd_n | psd_s]
    for (int c = 0; c < 6; c++)
      for (int e = 0; e < 6; e++) {
        float nre = sm[OFF_PRED + 256 + pidx(c, e)]     + sm[OFF_PRED + 256 + pidx(c + 6, e + 6)];
        float nim = sm[OFF_PRED + 256 + pidx(c + 6, e)] - sm[OFF_PRED + 256 + pidx(c, e + 6)];
        float sre = sm[OFF_PRED + pidx(c, e)]           + sm[OFF_PRED + pidx(c + 6, e + 6)];
        float sim = sm[OFF_PRED + pidx(c + 6, e)]       - sm[OFF_PRED + pidx(c, e + 6)];
        Ar[c][e] = nre;      Ai[c][e] = nim;
        Ar[c][6 + e] = sre;  Ai[c][6 + e] = sim;
      }
    // Tikhonov: psd_n += (1e-7*trace + 1e-8) * I
    float trn = 0.f;
    for (int c = 0; c < 6; c++) trn += Ar[c][c];
    float dl = trn * 1e-7f + 1e-8f;
    for (int c = 0; c < 6; c++) Ar[c][c] += dl;
    // Gauss-Jordan with partial pivoting
    for (int k = 0; k < 6; k++) {
      int p = k; float best = Ar[k][k] * Ar[k][k] + Ai[k][k] * Ai[k][k];
      for (int r = k + 1; r < 6; r++) {
        float m2 = Ar[r][k] * Ar[r][k] + Ai[r][k] * Ai[r][k];
        if (m2 > best) { best = m2; p = r; }
      }
      if (p != k)
        for (int col = 0; col < 12; col++) {
          float t0 = Ar[k][col]; Ar[k][col] = Ar[p][col]; Ar[p][col] = t0;
          float t1 = Ai[k][col]; Ai[k][col] = Ai[p][col]; Ai[p][col] = t1;
        }
      float d  = Ar[k][k] * Ar[k][k] + Ai[k][k] * Ai[k][k] + 1e-30f;
      float pr = Ar[k][k] / d, pi_ = -Ai[k][k] / d;
      for (int col = 0; col < 12; col++) {
        float a0 = Ar[k][col], b0 = Ai[k][col];
        Ar[k][col] = a0 * pr - b0 * pi_;
        Ai[k][col] = a0 * pi_ + b0 * pr;
      }
      for (int r = 0; r < 6; r++) {
        if (r == k) continue;
        float fr = Ar[r][k], fi = Ai[r][k];
        for (int col = 0; col < 12; col++) {
          float a0 = Ar[k][col], b0 = Ai[k][col];
          Ar[r][col] -= fr * a0 - fi * b0;
          Ai[r][col] -= fr * b0 + fi * a0;
        }
      }
    }
    // trace of numerator, w = N[:,0] / (tr + 1e-8)
    float trr = 1e-8f, tri = 0.f;
    for (int i = 0; i < 6; i++) { trr += Ar[i][6 + i]; tri += Ai[i][6 + i]; }
    float d2 = trr * trr + tri * tri + 1e-30f;
    for (int c = 0; c < 6; c++) {
      float nr = Ar[c][6], ni = Ai[c][6];
      sm[OFF_W + 2 * c]     = (nr * trr + ni * tri) / d2;
      sm[OFF_W + 2 * c + 1] = (ni * trr - nr * tri) / d2;
    }
  }
  __syncthreads();

  // ---------------- Phase D: beamform from the LDS copy of X
  float wr[6], wi[6];
#pragma unroll
  for (int c = 0; c < 6; c++) { wr[c] = sm[OFF_W + 2 * c]; wi[c] = sm[OFF_W + 2 * c + 1]; }
  float2* ob = (float2*)(out + (size_t)bf * Tn * 2);
  for (int t = tid; t < Tn; t += 256) {
    float ore = 0.f, oim = 0.f;
#pragma unroll
    for (int c = 0; c < 6; c++) {
      float xr = sm[OFF_X + c * XSTRIDE + t];
      float xi = sm[OFF_X + (6 + c) * XSTRIDE + t];
      ore += wr[c] * xr + wi[c] * xi;    // conj(w) * x
      oim += wr[c] * xi - wi[c] * xr;
    }
    ob[t] = make_float2(ore, oim);
  }
}

extern "C" void kernel_launch(void* const* d_in, const int* in_sizes, int n_in,
                              void* d_out, int out_size, void* d_ws, size_t ws_size,
                              hipStream_t stream) {
  const float* sr = (const float*)d_in[0];   // spec_real (B,F,C,T)
  const float* si = (const float*)d_in[1];   // spec_imag (B,F,C,T)
  const float* ms = (const float*)d_in[2];   // mask_s    (B,F,T)
  const float* mn = (const float*)d_in[3];   // mask_n    (B,F,T)
  float* out = (float*)d_out;                // complex64 (B,F,T) -> interleaved re,im

  dim3 grid(Bn * Fn);                        // one workgroup per (b,f)
  mvdr_kernel<<<grid, 256, 0, stream>>>(sr, si, ms, mn, out);
}